// ObsBCELoss_45956150067425
// MI455X (gfx1250) — compile-verified
//
#include <hip/hip_runtime.h>
#include <stdint.h>

// ---------------------------------------------------------------------------
// MI455X (gfx1250) implementation.
// Memory-bound streaming reduction (512MB logits+gumbel @ 23.3TB/s ~ 23us).
// CDNA5 path used: Tensor Data Mover (TENSOR_LOAD_TO_LDS) async DMA of
// 16KB tiles into LDS + s_wait_tensorcnt, wave32 shuffle reductions.
// Toolchain: amdgpu-toolchain (clang-23) -> 6-arg tensor_load_to_lds builtin.
// ---------------------------------------------------------------------------

typedef __attribute__((ext_vector_type(4))) unsigned int v4u;
typedef __attribute__((ext_vector_type(8))) int          v8i;
typedef __attribute__((ext_vector_type(4))) int          v4i;

#define BIN      256
#define ROWS_PB  16          // rows (columns-of-the-problem) per block
#define THREADS  256         // 8 waves of 32

// Build a D# for a contiguous 1-D copy of `nelem` float32 elements from
// global `gaddr` into LDS byte offset `lds_off`, and issue TENSOR_LOAD_TO_LDS.
// D# layout per CDNA5 ISA ch.8:
//   group0: [1:0]=count=1, [63:32]=lds_addr, [120:64]=global_addr, [127:126]=type=2
//   group1: [17:16]=data_size=2 (4B), tensor_dim0=[79:48], tensor_dim1=[111:80]=1,
//           tile_dim0=[127:112], tile_dim1=[143:128]=1, tensor_dim0_stride=[207:160]
__device__ __forceinline__ void tdm_load_1d_f32(uint32_t lds_off, uint64_t gaddr,
                                                uint32_t nelem) {
  v4u g0;
  g0[0] = 1u;                                        // count=1, user descriptor
  g0[1] = lds_off;                                   // LDS byte address
  g0[2] = (uint32_t)gaddr;                           // global_addr[31:0]
  g0[3] = ((uint32_t)(gaddr >> 32) & 0x01FFFFFFu)    // global_addr[56:32]
          | 0x80000000u;                             // type=2 ("image")
  v8i g1;
  g1[0] = 0x00020000;                                // data_size=2 -> 4 bytes
  g1[1] = (int)((nelem & 0xFFFFu) << 16);            // tensor_dim0[15:0]
  g1[2] = (int)(((nelem >> 16) & 0xFFFFu)            // tensor_dim0[31:16]
                | (1u << 16));                       // tensor_dim1 = 1
  g1[3] = (int)((nelem & 0xFFFFu) << 16);            // tile_dim0 = nelem
  g1[4] = 1;                                         // tile_dim1 = 1
  g1[5] = (int)nelem;                                // tensor_dim0_stride[31:0]
  g1[6] = 0;
  g1[7] = 0;
  v4i gz4 = {0, 0, 0, 0};                            // groups 2/3 unused (<=2D)
  v8i gz8 = {0, 0, 0, 0, 0, 0, 0, 0};                // trailing group (clang-23 form)
  __builtin_amdgcn_tensor_load_to_lds(g0, g1, gz4, gz4, gz8, /*cpol=*/0);
}

// One block: TDM-load a 16-row tile of logits+gumbel (2 x 16KB) into LDS,
// then 8 waves each reduce 2 rows (256 bins each) with wave32 shuffles.
__global__ void __launch_bounds__(THREADS)
col_nll_kernel(const float* __restrict__ logits,
               const float* __restrict__ gumbel,
               const unsigned char* __restrict__ mask,
               const int* __restrict__ targets,
               float* __restrict__ partials) {
  __shared__ __align__(16) float smem[2 * ROWS_PB * BIN];   // 32 KB
  __shared__ float red[ROWS_PB];

  const int tid  = threadIdx.x;
  const int w    = tid >> 5;
  const int lane = tid & 31;
  const uint64_t tile_elems = (uint64_t)ROWS_PB * BIN;      // 4096 floats
  const uint64_t base_elem  = (uint64_t)blockIdx.x * tile_elems;

  if (w == 0) {
    // Wave 0 issues both DMA tiles; TDM ignores EXEC, issued once per wave.
    uint32_t lds0 = (uint32_t)(uintptr_t)&smem[0];          // flat lo32 == LDS off
    tdm_load_1d_f32(lds0,                       (uint64_t)(uintptr_t)(logits + base_elem),
                    (uint32_t)tile_elems);
    tdm_load_1d_f32(lds0 + ROWS_PB * BIN * 4,   (uint64_t)(uintptr_t)(gumbel + base_elem),
                    (uint32_t)tile_elems);
    __builtin_amdgcn_s_wait_tensorcnt(0);                   // DMA complete
  }
  __syncthreads();                                          // publish LDS to all waves

#pragma unroll
  for (int it = 0; it < ROWS_PB / 8; ++it) {
    const int rl  = w + 8 * it;                             // local row 0..15
    const int row = blockIdx.x * ROWS_PB + rl;              // global (b*64+col)

    // Each lane owns 8 consecutive bins: two b128 LDS loads per operand.
    const float4* Lp = reinterpret_cast<const float4*>(&smem[rl * BIN]);
    const float4* Gp = reinterpret_cast<const float4*>(&smem[ROWS_PB * BIN + rl * BIN]);
    float4 l0 = Lp[lane * 2], l1 = Lp[lane * 2 + 1];
    float4 q0 = Gp[lane * 2], q1 = Gp[lane * 2 + 1];

    float z[8];
    z[0] = l0.x + q0.x; z[1] = l0.y + q0.y; z[2] = l0.z + q0.z; z[3] = l0.w + q0.w;
    z[4] = l1.x + q1.x; z[5] = l1.y + q1.y; z[6] = l1.z + q1.z; z[7] = l1.w + q1.w;

    // Row max (numerical stability), wave32 reduction.
    float m = z[0];
#pragma unroll
    for (int i = 1; i < 8; ++i) m = fmaxf(m, z[i]);
#pragma unroll
    for (int off = 16; off > 0; off >>= 1) m = fmaxf(m, __shfl_xor(m, off, 32));

    // Sum of exp(z - m), wave32 reduction.
    float s = 0.0f;
#pragma unroll
    for (int i = 0; i < 8; ++i) s += expf(z[i] - m);
#pragma unroll
    for (int off = 16; off > 0; off >>= 1) s += __shfl_xor(s, off, 32);

    // Target pick: t uniform across the wave (one row per wave).
    const int t  = targets[row];
    const int i0 = t & 7;
    float zsel = z[0];
#pragma unroll
    for (int i = 1; i < 8; ++i) zsel = (i0 == i) ? z[i] : zsel;
    const float zt  = __shfl(zsel, t >> 3, 32);
    const float lse = m + logf(s);

    const unsigned char mk = mask[(size_t)row * BIN + t];   // single byte per row
    const float contrib = mk ? 0.0f : (lse - zt);           // = -picked

    if (lane == 0) red[rl] = contrib;
  }
  __syncthreads();

  if (tid == 0) {
    float acc = 0.0f;
#pragma unroll
    for (int i = 0; i < ROWS_PB; ++i) acc += red[i];
    partials[blockIdx.x] = acc;
  }
}

// Deterministic second-stage reduction of per-block partials.
__global__ void __launch_bounds__(256)
final_reduce_kernel(const float* __restrict__ partials, int n, int batch,
                    float* __restrict__ out) {
  __shared__ float sm[256];
  float acc = 0.0f;
  for (int i = threadIdx.x; i < n; i += 256) acc += partials[i];
  sm[threadIdx.x] = acc;
  __syncthreads();
#pragma unroll
  for (int s = 128; s > 0; s >>= 1) {
    if ((int)threadIdx.x < s) sm[threadIdx.x] += sm[threadIdx.x + s];
    __syncthreads();
  }
  if (threadIdx.x == 0) {
    const float loss = sm[0];
    out[0] = loss;
    out[1] = loss / ((float)batch * 0.69314718055994530942f);  // loss / (B*ln2)
  }
}

extern "C" void kernel_launch(void* const* d_in, const int* in_sizes, int n_in,
                              void* d_out, int out_size, void* d_ws, size_t ws_size,
                              hipStream_t stream) {
  const float*         logits  = (const float*)d_in[0];
  const float*         gumbel  = (const float*)d_in[1];
  const unsigned char* mask    = (const unsigned char*)d_in[2];
  const int*           targets = (const int*)d_in[3];
  float*               out     = (float*)d_out;
  float*               parts   = (float*)d_ws;   // nb floats (64 KB) of scratch

  const int rows = in_sizes[3];        // B * N_COLS = 262144
  const int nb   = rows / ROWS_PB;     // 16384 blocks
  const int b    = rows / 64;          // B = 4096

  col_nll_kernel<<<nb, THREADS, 0, stream>>>(logits, gumbel, mask, targets, parts);
  final_reduce_kernel<<<1, 256, 0, stream>>>(parts, nb, b, out);
}